// SceneFlowLoss_40776419508561
// MI455X (gfx1250) — compile-verified
//
#include <hip/hip_runtime.h>
#include <hip/hip_bf16.h>
#include <math.h>

typedef __attribute__((ext_vector_type(2))) float v2f;
typedef __attribute__((ext_vector_type(8))) float v8f;

#define W_CHAMFER 1.0f
#define W_SMOOTH  0.5f
#define KNN       8

// Sum values held by lanes 0..15 across the wave (wave32); result in all lanes.
__device__ __forceinline__ float wave_sum_lo16(float v, int lane) {
    if (lane >= 16) v = 0.0f;
    v += __shfl_xor(v, 1, 32);
    v += __shfl_xor(v, 2, 32);
    v += __shfl_xor(v, 4, 32);
    v += __shfl_xor(v, 8, 32);
    v += __shfl_xor(v, 16, 32);
    return v;
}

// ---------------------------------------------------------------------------
// pc1_warped = pc1 + flow (precomputed once; removes flow math from hot loops)
// ---------------------------------------------------------------------------
__global__ void warp_kernel(const float* __restrict__ p,
                            const float* __restrict__ f,
                            float* __restrict__ o, int n) {
    const int i = blockIdx.x * blockDim.x + threadIdx.x;
    if (i < n) o[i] = p[i] + f[i];
}

// ---------------------------------------------------------------------------
// Chamfer direction: per query (16 per wave-chunk, QC=2 chunks per wave),
// min squared distance to all candidates via V_WMMA_F32_16X16X4_F32 tiles:
//   A (cand x K)  = [cx, cy, cz, |c|^2]
//   B (K x query) = [-2qx, -2qy, -2qz, 1]
//   D[c,q]        = |c|^2 - 2 q.c        (+|q|^2 per-lane constant at the end)
// QF/CF template flags fold optional flow adds with no inner-loop branching.
// Candidate tiles are double-buffered in registers (prefetch next, use prev).
// ---------------------------------------------------------------------------
template <bool QF, bool CF>
__global__ __launch_bounds__(256) void chamfer_min_kernel(
    const float* __restrict__ qp, const float* __restrict__ qflow,
    const float* __restrict__ cp, const float* __restrict__ cflow,
    float* __restrict__ accum, int nq, int nc)
{
    constexpr int QC = 2;                      // query chunks per wave
    const int lane  = threadIdx.x & 31;
    const int wave  = (blockIdx.x * blockDim.x + threadIdx.x) >> 5;
    const int groupsPerBatch = (nq >> 4) / QC;
    const int b     = wave / groupsPerBatch;
    const int g     = wave % groupsPerBatch;
    const int j     = lane & 15;
    const bool hi   = lane >= 16;

    const float* qb  = qp + (size_t)b * nq * 3;
    const float* cb  = cp + (size_t)b * nc * 3;
    const float* qfb = QF ? (qflow + (size_t)b * nq * 3) : nullptr;
    const float* cfb = CF ? (cflow + (size_t)b * nc * 3) : nullptr;

    v2f   bvec[QC];
    float q2[QC], best[QC];
#pragma unroll
    for (int c = 0; c < QC; ++c) {
        const int qi = (g * QC + c) * 16 + j;
        float qx = qb[3 * qi + 0], qy = qb[3 * qi + 1], qz = qb[3 * qi + 2];
        if (QF) { qx += qfb[3 * qi + 0]; qy += qfb[3 * qi + 1]; qz += qfb[3 * qi + 2]; }
        q2[c] = qx * qx + qy * qy + qz * qz;
        bvec[c].x = hi ? (-2.0f * qz) : (-2.0f * qx);   // K=2 : K=0
        bvec[c].y = hi ? 1.0f         : (-2.0f * qy);   // K=3 : K=1
        best[c] = 3.4e38f;
    }

    // Prime the register double-buffer with tile 0.
    float cx = cb[3 * j + 0], cy = cb[3 * j + 1], cz = cb[3 * j + 2];
    if (CF) { cx += cfb[3 * j + 0]; cy += cfb[3 * j + 1]; cz += cfb[3 * j + 2]; }

    for (int m0 = 0; m0 < nc; m0 += 16) {
        // Issue next tile's loads early (clamped on the last iteration).
        const int nb  = (m0 + 16 < nc) ? (m0 + 16) : 0;
        const int nci = nb + j;
        float nx = cb[3 * nci + 0], ny = cb[3 * nci + 1], nz = cb[3 * nci + 2];
        if (CF) { nx += cfb[3 * nci + 0]; ny += cfb[3 * nci + 1]; nz += cfb[3 * nci + 2]; }

        const float c2 = cx * cx + cy * cy + cz * cz;
        v2f avec;
        avec.x = hi ? cz : cx;   // K=2 : K=0
        avec.y = hi ? c2 : cy;   // K=3 : K=1

#pragma unroll
        for (int c = 0; c < QC; ++c) {
            v8f acc = {};
            acc = __builtin_amdgcn_wmma_f32_16x16x4_f32(
                false, avec, false, bvec[c], (short)0, acc, false, false);
#pragma unroll
            for (int r = 0; r < 8; ++r) best[c] = fminf(best[c], acc[r]);
        }

        cx = nx; cy = ny; cz = nz;
    }

    float s = 0.0f;
#pragma unroll
    for (int c = 0; c < QC; ++c) {
        const float m = fminf(best[c], __shfl_xor(best[c], 16, 32));
        s += sqrtf(fmaxf(m + q2[c], 0.0f));
    }
    s = wave_sum_lo16(s, lane);
    if (lane == 0) atomicAdd(accum, s);
}

// ---------------------------------------------------------------------------
// KNN smoothness: 8-NN of pc1 vs itself. Each lane tracks a sorted 9-deep
// (score, idx) register list over ITS OWN M-half only (no shuffles in the hot
// loop). Candidate tiles are double-buffered in registers so the memory wait
// overlaps the WMMA + insertion work. At the end, a single bitonic-crossover
// merge with the partner lane (min(A[k], B[8-k])) yields the 9 global-
// smallest; the self match is dropped by index. |q|^2 is a per-lane constant,
// so raw WMMA scores preserve ordering.
// ---------------------------------------------------------------------------
__global__ __launch_bounds__(256) void knn_smooth_kernel(
    const float* __restrict__ pts, const float* __restrict__ flow,
    float* __restrict__ accum, int n)
{
    const int lane  = threadIdx.x & 31;
    const int wave  = (blockIdx.x * blockDim.x + threadIdx.x) >> 5;
    const int chunksPerBatch = n >> 4;
    const int b     = wave / chunksPerBatch;
    const int chunk = wave % chunksPerBatch;
    const int j     = lane & 15;
    const bool hi   = lane >= 16;

    const float* pb = pts  + (size_t)b * n * 3;
    const float* fb = flow + (size_t)b * n * 3;

    const int qi = chunk * 16 + j;
    const float qx = pb[3 * qi + 0], qy = pb[3 * qi + 1], qz = pb[3 * qi + 2];

    v2f bvec;
    bvec.x = hi ? (-2.0f * qz) : (-2.0f * qx);
    bvec.y = hi ? 1.0f         : (-2.0f * qy);

    float bd[KNN + 1];
    int   bi[KNN + 1];
#pragma unroll
    for (int k = 0; k <= KNN; ++k) { bd[k] = 3.4e38f; bi[k] = -1; }

#define INSERT(vv, ii)                                                        \
    do {                                                                      \
        float _v = (vv); int _i = (ii);                                       \
        if (_v < bd[KNN]) {                                                   \
            bd[KNN] = _v; bi[KNN] = _i;                                       \
            _Pragma("unroll")                                                 \
            for (int _k = KNN; _k > 0; --_k) {                                \
                if (bd[_k] < bd[_k - 1]) {                                    \
                    float _td = bd[_k]; bd[_k] = bd[_k - 1]; bd[_k - 1] = _td;\
                    int   _ti = bi[_k]; bi[_k] = bi[_k - 1]; bi[_k - 1] = _ti;\
                }                                                             \
            }                                                                 \
        }                                                                     \
    } while (0)

    const int mofs = hi ? 8 : 0;   // own M-half offset within the tile

    // Prime the register double-buffer with tile 0.
    float cx = pb[3 * j + 0], cy = pb[3 * j + 1], cz = pb[3 * j + 2];

    for (int m0 = 0; m0 < n; m0 += 16) {
        // Issue next tile's load early (clamped on the last iteration) so the
        // wait overlaps this tile's WMMA + insertion work.
        const int nci = ((m0 + 16 < n) ? (m0 + 16) : 0) + j;
        const float nx = pb[3 * nci + 0];
        const float ny = pb[3 * nci + 1];
        const float nz = pb[3 * nci + 2];

        const float c2 = cx * cx + cy * cy + cz * cz;
        v2f avec;
        avec.x = hi ? cz : cx;
        avec.y = hi ? c2 : cy;

        v8f acc = {};
        acc = __builtin_amdgcn_wmma_f32_16x16x4_f32(
            false, avec, false, bvec, (short)0, acc, false, false);

#pragma unroll
        for (int r = 0; r < 8; ++r) INSERT(acc[r], m0 + r + mofs);

        cx = nx; cy = ny; cz = nz;
    }
#undef INSERT

    // Bitonic crossover with partner lane: both ascending 9-lists; the 9
    // smallest of the union are min(A[k], B[8-k]) (order not needed further).
#pragma unroll
    for (int k = 0; k <= KNN; ++k) {
        const float pd = __shfl_xor(bd[KNN - k], 16, 32);
        const int   pi = __shfl_xor(bi[KNN - k], 16, 32);
        if (pd < bd[k]) { bd[k] = pd; bi[k] = pi; }
    }

    float s = 0.0f;
    if (!hi) {
        const float fx = fb[3 * qi + 0], fy = fb[3 * qi + 1], fz = fb[3 * qi + 2];
#pragma unroll
        for (int k = 0; k <= KNN; ++k) {
            const int t = bi[k];
            if (t != qi) {               // drop the self match (exactly one)
                const float dx = fx - fb[3 * t + 0];
                const float dy = fy - fb[3 * t + 1];
                const float dz = fz - fb[3 * t + 2];
                s += sqrtf(dx * dx + dy * dy + dz * dz);
            }
        }
    }
    s = wave_sum_lo16(s, lane);
    if (lane == 0) atomicAdd(accum, s);
}

__global__ void zero_ws_kernel(float* ws) {
    if (threadIdx.x < 3) ws[threadIdx.x] = 0.0f;
}

__global__ void finalize_kernel(const float* __restrict__ ws,
                                float* __restrict__ out,
                                int B, int N, int M)
{
    if (threadIdx.x == 0 && blockIdx.x == 0) {
        const float chamfer = ws[0] / (float)(B * N) + ws[1] / (float)(B * M);
        const float smooth  = ws[2] / (float)(B * N * KNN);
        out[0] = W_CHAMFER * chamfer + W_SMOOTH * smooth;
    }
}

extern "C" void kernel_launch(void* const* d_in, const int* in_sizes, int n_in,
                              void* d_out, int out_size, void* d_ws, size_t ws_size,
                              hipStream_t stream) {
    const float* pc1  = (const float*)d_in[0];
    const float* pc2  = (const float*)d_in[1];
    const float* flow = (const float*)d_in[2];
    float* ws  = (float*)d_ws;
    float* out = (float*)d_out;

    const int B = 2;                       // per reference setup
    const int N = in_sizes[0] / (B * 3);
    const int M = in_sizes[1] / (B * 3);

    zero_ws_kernel<<<1, 32, 0, stream>>>(ws);

    const int threads = 256;               // 8 waves/block
    constexpr int QC = 2;
    const int wavesQN = B * (N / 16) / QC;
    const int wavesQM = B * (M / 16) / QC;
    const int blocksQN = (wavesQN * 32 + threads - 1) / threads;
    const int blocksQM = (wavesQM * 32 + threads - 1) / threads;

    const int    n3        = B * N * 3;
    const size_t warpBytes = 256 + (size_t)n3 * sizeof(float);
    float* warped = (float*)((char*)d_ws + 256);

    if (ws_size >= warpBytes) {
        // Precompute pc1 + flow once; both chamfer directions run the
        // branch-free no-flow specialization.
        warp_kernel<<<(n3 + threads - 1) / threads, threads, 0, stream>>>(
            pc1, flow, warped, n3);
        chamfer_min_kernel<false, false><<<blocksQN, threads, 0, stream>>>(
            warped, nullptr, pc2, nullptr, ws + 0, N, M);
        chamfer_min_kernel<false, false><<<blocksQM, threads, 0, stream>>>(
            pc2, nullptr, warped, nullptr, ws + 1, M, N);
    } else {
        chamfer_min_kernel<true, false><<<blocksQN, threads, 0, stream>>>(
            pc1, flow, pc2, nullptr, ws + 0, N, M);
        chamfer_min_kernel<false, true><<<blocksQM, threads, 0, stream>>>(
            pc2, nullptr, pc1, flow, ws + 1, M, N);
    }

    // smoothness: 8-NN of pc1 vs itself
    const int wavesK  = B * (N / 16);
    const int blocksK = (wavesK * 32 + threads - 1) / threads;
    knn_smooth_kernel<<<blocksK, threads, 0, stream>>>(pc1, flow, ws + 2, N);

    finalize_kernel<<<1, 1, 0, stream>>>(ws, out, B, N, M);
}